// st_ode_68882685493267
// MI455X (gfx1250) — compile-verified
//
#include <hip/hip_runtime.h>

// ---------------- problem constants (from reference) ----------------
#define Bsz 32
#define Cch 64
#define Tt  256
#define Vv  62
#define EPSF 1e-5f
#define STEP 1.0f      // 0.5 * ALPHA, ALPHA = 2.0

#define LDP 68         // padded LDS row stride (floats) for column-strided reads
#define TT  4          // t-values per stage-B block
#define NS  (TT + 8)   // input slices needed (4-tap halo each side)

typedef float v2f __attribute__((ext_vector_type(2)));
typedef float v8f __attribute__((ext_vector_type(8)));

// ---------------------------------------------------------------
// k_prep: A_norm = D^-1/2 (adj + I) D^-1/2, zero-padded to 64x64
// ---------------------------------------------------------------
__global__ void k_prep(const float* __restrict__ adj, float* __restrict__ anorm) {
    __shared__ float dinv[64];
    int v = threadIdx.x;                      // 64 threads
    float di = 0.f;
    if (v < Vv) {
        float s = 1.0f;                       // self loop
        for (int w = 0; w < Vv; ++w) s += adj[v * Vv + w];
        di = rsqrtf(s);
    }
    dinv[v] = di;
    __syncthreads();
    for (int w = 0; w < 64; ++w) {
        float val = 0.f;
        if (v < Vv && w < Vv) {
            float a = adj[v * Vv + w] + (v == w ? 1.f : 0.f);
            val = a * dinv[v] * dinv[w];
        }
        anorm[v * 64 + w] = val;
    }
}

// ---------------------------------------------------------------
// Stage A: per (b,t): Y = W_mlp * ( X + STEP*(X*A - X) ) + b_mlp
// Two 64x64x64 GEMMs via v_wmma_f32_16x16x4_f32. Also accumulates
// per-channel sum / sumsq of Y for BN1 batch statistics.
// 256 threads = 8 waves; 16 output tiles of 16x16 -> 2 tiles/wave.
// ---------------------------------------------------------------
__global__ __launch_bounds__(256) void k_stageA(
    const float* __restrict__ x, const float* __restrict__ anorm,
    const float* __restrict__ wmlp, const float* __restrict__ bmlp,
    float* __restrict__ y1, float* __restrict__ stats1)
{
    extern __shared__ float smem[];
    float* sX   = smem;                 // 64*LDP  (X: rows c, cols v, zero padded)
    float* sA   = sX  + 64 * LDP;       // 64*LDP  (A_norm: rows w, cols v)
    float* sW   = sA  + 64 * LDP;       // 64*LDP  (W_mlp: rows o, cols c)
    float* sAX  = sW  + 64 * LDP;       // 64*LDP  (h = X + s*(XA - X))
    float* sSum = sAX + 64 * LDP;       // 64
    float* sSq  = sSum + 64;            // 64

    const int tid = threadIdx.x;
    const int b = blockIdx.x / Tt;
    const int t = blockIdx.x % Tt;

    if (tid < 64) { sSum[tid] = 0.f; sSq[tid] = 0.f; }

    for (int i = tid; i < 64 * 64; i += 256) {
        int r = i >> 6, c = i & 63;
        sA[r * LDP + c] = anorm[i];
        sW[r * LDP + c] = wmlp[i];
        float xv = 0.f;
        if (c < Vv) xv = x[((size_t)(b * Cch + r) * Tt + t) * Vv + c];
        sX[r * LDP + c] = xv;
    }
    __syncthreads();

    const int wave  = tid >> 5;
    const int lane  = tid & 31;
    const int l16   = lane & 15;
    const int hiOff = (lane >= 16) ? 2 : 0;   // K half selector (A/B frag)
    const int hiRow = (lane >= 16) ? 8 : 0;   // M half selector (C/D frag)

    // ---- GEMM 1: AX = X * A  (M=c, N=v, K=w), epilogue h = x + s*(ax-x)
    for (int i = 0; i < 2; ++i) {
        int tile = wave * 2 + i;
        int m0 = (tile >> 2) * 16, n0 = (tile & 3) * 16;
        v8f acc; for (int r = 0; r < 8; ++r) acc[r] = 0.f;
        for (int k = 0; k < 64; k += 4) {
            v2f a, bf;
            a.x  = sX[(m0 + l16) * LDP + k + hiOff];
            a.y  = sX[(m0 + l16) * LDP + k + hiOff + 1];
            bf.x = sA[(k + hiOff) * LDP + n0 + l16];
            bf.y = sA[(k + hiOff + 1) * LDP + n0 + l16];
            acc = __builtin_amdgcn_wmma_f32_16x16x4_f32(false, a, false, bf,
                                                        (short)0, acc, false, false);
        }
        for (int r = 0; r < 8; ++r) {
            int m = m0 + r + hiRow, n = n0 + l16;
            float xv = sX[m * LDP + n];
            sAX[m * LDP + n] = xv + STEP * (acc[r] - xv);
        }
    }
    __syncthreads();

    // ---- GEMM 2: Y = W * h + b_mlp  (M=o, N=v, K=c)
    for (int i = 0; i < 2; ++i) {
        int tile = wave * 2 + i;
        int m0 = (tile >> 2) * 16, n0 = (tile & 3) * 16;
        v8f acc; for (int r = 0; r < 8; ++r) acc[r] = 0.f;
        for (int k = 0; k < 64; k += 4) {
            v2f a, bf;
            a.x  = sW[(m0 + l16) * LDP + k + hiOff];
            a.y  = sW[(m0 + l16) * LDP + k + hiOff + 1];
            bf.x = sAX[(k + hiOff) * LDP + n0 + l16];
            bf.y = sAX[(k + hiOff + 1) * LDP + n0 + l16];
            acc = __builtin_amdgcn_wmma_f32_16x16x4_f32(false, a, false, bf,
                                                        (short)0, acc, false, false);
        }
        int n = n0 + l16;
        if (n < Vv) {
            for (int r = 0; r < 8; ++r) {
                int o = m0 + r + hiRow;
                float yv = acc[r] + bmlp[o];
                y1[((size_t)(b * Cch + o) * Tt + t) * Vv + n] = yv;
                atomicAdd(&sSum[o], yv);
                atomicAdd(&sSq[o], yv * yv);
            }
        }
    }
    __syncthreads();
    if (tid < 64) {
        atomicAdd(&stats1[tid], sSum[tid]);
        atomicAdd(&stats1[64 + tid], sSq[tid]);
    }
}

// ---------------------------------------------------------------
// finalize BN stats -> per-channel scale/shift (fold gamma/beta)
// ---------------------------------------------------------------
__global__ void k_finalize(const float* __restrict__ stats,
                           const float* __restrict__ g, const float* __restrict__ bb,
                           float* __restrict__ scale, float* __restrict__ shift, float n)
{
    int c = threadIdx.x;
    if (c < Cch) {
        float mean = stats[c] / n;
        float var  = stats[64 + c] / n - mean * mean;
        float rstd = rsqrtf(var + EPSF);
        float sc = g[c] * rstd;
        scale[c] = sc;
        shift[c] = bb[c] - mean * sc;
    }
}

// ---------------------------------------------------------------
// Stage B: temporal conv as GEMM. Block = (b, 4 consecutive t).
// LDS holds 12 BN1+ReLU'd [C x 64] slices (t0-4 .. t0+7) + one tap
// of W (64x64). For each tap k: Y_t += Wk * Z_{t+k-4}. 64 output
// tiles / 8 waves = 8 accumulator tiles per wave.
// Writes y2 (staged in d_out) + BN2 batch statistics.
// ---------------------------------------------------------------
__global__ __launch_bounds__(256) void k_stageB(
    const float* __restrict__ y1, const float* __restrict__ wconv,
    const float* __restrict__ bconv,
    const float* __restrict__ sc1, const float* __restrict__ sh1,
    float* __restrict__ y2, float* __restrict__ stats2)
{
    extern __shared__ float smem[];
    float* zbuf = smem;                    // NS * 64 * 64
    float* sWk  = zbuf + NS * 64 * 64;     // 64 * LDP
    float* sSum = sWk + 64 * LDP;          // 64
    float* sSq  = sSum + 64;               // 64

    const int tid = threadIdx.x;
    const int b  = blockIdx.x / (Tt / TT);
    const int t0 = (blockIdx.x % (Tt / TT)) * TT;

    if (tid < 64) { sSum[tid] = 0.f; sSq[tid] = 0.f; }

    // load slices with BN1 + ReLU applied on the fly (zero pad t & v)
    for (int i = tid; i < NS * 64 * 64; i += 256) {
        int s = i >> 12;            // slice 0..11
        int c = (i >> 6) & 63;
        int v = i & 63;
        int tin = t0 - 4 + s;
        float z = 0.f;
        if (v < Vv && tin >= 0 && tin < Tt) {
            float yv = y1[((size_t)(b * Cch + c) * Tt + tin) * Vv + v];
            z = yv * sc1[c] + sh1[c];
            z = z > 0.f ? z : 0.f;
        }
        zbuf[i] = z;
    }

    const int wave  = tid >> 5;
    const int lane  = tid & 31;
    const int l16   = lane & 15;
    const int hiOff = (lane >= 16) ? 2 : 0;
    const int hiRow = (lane >= 16) ? 8 : 0;

    v8f acc[8];
    for (int j = 0; j < 8; ++j)
        for (int r = 0; r < 8; ++r) acc[j][r] = 0.f;

    for (int k = 0; k < 9; ++k) {
        __syncthreads();   // zbuf ready (k==0) / prev tap's sWk consumers done
        for (int i = tid; i < 64 * 64; i += 256) {
            int o = i >> 6, c = i & 63;
            sWk[o * LDP + c] = wconv[(size_t)(o * Cch + c) * 9 + k];
        }
        __syncthreads();
        for (int j = 0; j < 8; ++j) {
            int tile = wave * 8 + j;            // 0..63
            int tl = tile >> 4;                 // t_local 0..3
            int m0 = ((tile >> 2) & 3) * 16;
            int n0 = (tile & 3) * 16;
            const float* zs = zbuf + (tl + k) * 4096;
            v8f a8 = acc[j];
            for (int kk = 0; kk < 64; kk += 4) {
                v2f a, bf;
                a.x  = sWk[(m0 + l16) * LDP + kk + hiOff];
                a.y  = sWk[(m0 + l16) * LDP + kk + hiOff + 1];
                bf.x = zs[(kk + hiOff) * 64 + n0 + l16];
                bf.y = zs[(kk + hiOff + 1) * 64 + n0 + l16];
                a8 = __builtin_amdgcn_wmma_f32_16x16x4_f32(false, a, false, bf,
                                                           (short)0, a8, false, false);
            }
            acc[j] = a8;
        }
    }

    for (int j = 0; j < 8; ++j) {
        int tile = wave * 8 + j;
        int tl = tile >> 4;
        int m0 = ((tile >> 2) & 3) * 16;
        int n  = (tile & 3) * 16 + l16;
        if (n < Vv) {
            int t = t0 + tl;
            for (int r = 0; r < 8; ++r) {
                int o = m0 + r + hiRow;
                float yv = acc[j][r] + bconv[o];
                y2[((size_t)(b * Cch + o) * Tt + t) * Vv + n] = yv;
                atomicAdd(&sSum[o], yv);
                atomicAdd(&sSq[o], yv * yv);
            }
        }
    }
    __syncthreads();
    if (tid < 64) {
        atomicAdd(&stats2[tid], sSum[tid]);
        atomicAdd(&stats2[64 + tid], sSq[tid]);
    }
}

// ---------------------------------------------------------------
// Final: out = relu( y2*scale2 + shift2 + x )   (in-place on d_out)
// ---------------------------------------------------------------
__global__ void k_final(const float* __restrict__ y2, const float* __restrict__ x,
                        const float* __restrict__ sc2, const float* __restrict__ sh2,
                        float* __restrict__ out, int n)
{
    int i = blockIdx.x * 256 + threadIdx.x;
    if (i < n) {
        int c = (i / (Tt * Vv)) % Cch;
        float v = y2[i] * sc2[c] + sh2[c] + x[i];
        out[i] = v > 0.f ? v : 0.f;
    }
}

// ---------------------------------------------------------------
extern "C" void kernel_launch(void* const* d_in, const int* in_sizes, int n_in,
                              void* d_out, int out_size, void* d_ws, size_t ws_size,
                              hipStream_t stream) {
    (void)in_sizes; (void)n_in; (void)ws_size;
    const float* x     = (const float*)d_in[0];
    const float* adj   = (const float*)d_in[1];
    const float* wmlp  = (const float*)d_in[2];
    const float* bmlp  = (const float*)d_in[3];
    const float* g1    = (const float*)d_in[4];
    const float* b1    = (const float*)d_in[5];
    const float* wconv = (const float*)d_in[6];
    const float* bconv = (const float*)d_in[7];
    const float* g2    = (const float*)d_in[8];
    const float* b2    = (const float*)d_in[9];
    float* out = (float*)d_out;

    // workspace layout
    char* ws = (char*)d_ws;
    const size_t y1_bytes = (size_t)Bsz * Cch * Tt * Vv * sizeof(float);  // ~130 MB
    float* y1     = (float*)ws;
    float* anorm  = (float*)(ws + y1_bytes);   // 64*64
    float* stats1 = anorm + 64 * 64;           // 128 (sum, sumsq)
    float* stats2 = stats1 + 128;              // 128
    float* sc1    = stats2 + 128;              // 64
    float* sh1    = sc1 + 64;                  // 64
    float* sc2    = sh1 + 64;                  // 64
    float* sh2    = sc2 + 64;                  // 64

    hipMemsetAsync(stats1, 0, 256 * sizeof(float), stream);

    k_prep<<<1, 64, 0, stream>>>(adj, anorm);

    size_t shmA = (size_t)(4 * 64 * LDP + 128) * sizeof(float);      // ~70 KB
    k_stageA<<<Bsz * Tt, 256, shmA, stream>>>(x, anorm, wmlp, bmlp, y1, stats1);

    float n1 = (float)((size_t)Bsz * Tt * Vv);
    k_finalize<<<1, 64, 0, stream>>>(stats1, g1, b1, sc1, sh1, n1);

    size_t shmB = ((size_t)NS * 64 * 64 + 64 * LDP + 128) * sizeof(float);  // ~210 KB
    k_stageB<<<Bsz * (Tt / TT), 256, shmB, stream>>>(y1, wconv, bconv, sc1, sh1,
                                                     out, stats2);

    k_finalize<<<1, 64, 0, stream>>>(stats2, g2, b2, sc2, sh2, n1);

    int n = Bsz * Cch * Tt * Vv;
    k_final<<<(n + 255) / 256, 256, 0, stream>>>(out, x, sc2, sh2, out, n);
}